// WL_58059367907459
// MI455X (gfx1250) — compile-verified
//
#include <hip/hip_runtime.h>
#include <stdint.h>

// Weisfeiler-Lehman color refinement, 3 iterations, exact reproduction of the
// reference's sorted-unique relabeling. Memory-bound: ~614 MB of edge stream
// over 3 iters (~26us floor at 23.3 TB/s HBM); tables are L2-resident (192MB L2).
// Edge stream is staged through LDS with gfx1250 async global->LDS copies
// (ASYNCcnt) and double buffering to hide HBM latency behind the gather+atomic
// dependency chain.

typedef long long          i64;
typedef unsigned long long u64;

typedef int v4i __attribute__((ext_vector_type(4)));
typedef __attribute__((address_space(1))) v4i gv4i;   // global int4
typedef __attribute__((address_space(3))) v4i lv4i;   // LDS int4

#define BLK    256          // 8 wave32 waves per block
#define CHUNK  2048         // elements per block in sort/scan kernels
#define TILES  (CHUNK / BLK)
#define NUM_IT 3
#define MASK40 ((1ull << 40) - 1ull)

#define EPT    2            // edges per thread per tile (16B = one async b128)
#define TILE_E (BLK * EPT)  // 512 edges per tile

static __device__ __forceinline__ int imin(int a, int b) { return a < b ? a : b; }

// ---------------------------------------------------------------- init / clear
__global__ void wl_init_colors(const i64* __restrict__ x, i64* __restrict__ colors, int n) {
  int i = blockIdx.x * blockDim.x + threadIdx.x;
  if (i < n) colors[i] = x[i];
}

__global__ void wl_clear_msum(u64* __restrict__ msum, int n) {
  int i = blockIdx.x * blockDim.x + threadIdx.x;
  if (i < n) msum[i] = 0ull;
}

// -------------------------------------------------------------- edge scatter
// Per edge: msum[dst] += hash_table[colors[src]]  (u64 wrap == int64 wrap; the
// final &MASK40 matches the reference's floor-mod exactly since 2^40 | 2^64).
// Full tiles: async global->LDS (per-lane b128, double-buffered, ASYNCcnt
// waits). Tail / misaligned: direct bounds-checked loads.
__global__ void wl_edge_scatter(const i64* __restrict__ src,
                                const i64* __restrict__ dst,
                                const i64* __restrict__ colors,
                                const i64* __restrict__ ht,
                                u64* __restrict__ msum,
                                long long E, long long nTiles, int asyncOK) {
  __shared__ __align__(16) i64 lsrc[2][TILE_E];
  __shared__ __align__(16) i64 ldst[2][TILE_E];
  int tid = threadIdx.x;

  auto issue = [&](long long tile, int buf) {
    long long base = tile * (long long)TILE_E + (long long)tid * EPT;
    // per-lane 16B async copies, tracked by ASYNCcnt (2 per wave per tile)
    __builtin_amdgcn_global_load_async_to_lds_b128(
        (gv4i*)(void*)(src + base), (lv4i*)&lsrc[buf][tid * EPT], 0, 0);
    __builtin_amdgcn_global_load_async_to_lds_b128(
        (gv4i*)(void*)(dst + base), (lv4i*)&ldst[buf][tid * EPT], 0, 0);
  };

  long long tile   = blockIdx.x;
  long long stride = gridDim.x;
  if (asyncOK && tile < nTiles && (tile + 1) * (long long)TILE_E <= E) issue(tile, 0);

  int buf = 0;
  for (; tile < nTiles; tile += stride, buf ^= 1) {
    __syncthreads();  // close WAR window: prior LDS reads done before reuse
    long long next  = tile + stride;
    bool nextFull   = asyncOK && (next < nTiles) && ((next + 1) * (long long)TILE_E <= E);
    bool curFull    = asyncOK && ((tile + 1) * (long long)TILE_E <= E);
    if (nextFull) issue(next, buf ^ 1);
    // keep the stream warm two tiles ahead in L2 (global_prefetch_b8)
    long long pf = (tile + 2 * stride) * (long long)TILE_E + (long long)tid * EPT;
    if (pf < E) {
      __builtin_prefetch(src + pf, 0, 1);
      __builtin_prefetch(dst + pf, 0, 1);
    }
    if (curFull) {
      // drain current tile's pair; next tile's pair may remain in flight
      if (nextFull) __builtin_amdgcn_s_wait_asynccnt(2);
      else          __builtin_amdgcn_s_wait_asynccnt(0);
      i64 s0 = lsrc[buf][tid * EPT + 0];
      i64 s1 = lsrc[buf][tid * EPT + 1];
      i64 d0 = ldst[buf][tid * EPT + 0];
      i64 d1 = ldst[buf][tid * EPT + 1];
      atomicAdd(&msum[d0], (u64)ht[colors[s0]]);
      atomicAdd(&msum[d1], (u64)ht[colors[s1]]);
    } else {
      for (int k = 0; k < EPT; ++k) {
        long long e = tile * (long long)TILE_E + (long long)tid * EPT + k;
        if (e < E) {
          i64 s = __builtin_nontemporal_load(src + e);
          i64 d = __builtin_nontemporal_load(dst + e);
          atomicAdd(&msum[d], (u64)ht[colors[s]]);
        }
      }
    }
  }
}

// ------------------------------------------------------------------ make keys
__global__ void wl_make_keys(const i64* __restrict__ colors,
                             const u64* __restrict__ msum,
                             u64* __restrict__ keyA, int* __restrict__ idxA, int n) {
  int i = blockIdx.x * blockDim.x + threadIdx.x;
  if (i < n) {
    keyA[i] = ((u64)colors[i] << 40) | (msum[i] & MASK40);
    idxA[i] = i;
  }
}

// ------------------------------------------------- radix sort: histogram pass
// hist layout is digit-major: hist[d * nBlocks + b]
__global__ void wl_radix_hist(const u64* __restrict__ keys, int n, int shift,
                              int* __restrict__ hist, int nBlocks) {
  __shared__ int h[256];
  int tid = threadIdx.x;
  h[tid] = 0;
  __syncthreads();
  int base = blockIdx.x * CHUNK;
  for (int t = 0; t < TILES; ++t) {
    int j = base + t * BLK + tid;
    if (j < n) atomicAdd(&h[(int)((keys[j] >> shift) & 255ull)], 1);
  }
  __syncthreads();
  hist[tid * nBlocks + blockIdx.x] = h[tid];
}

// ----------------------------------- generic single-block exclusive scan (m ints)
__global__ void wl_scan_exclusive(int* __restrict__ data, int m) {
  __shared__ int partial[BLK];
  int tid = threadIdx.x;
  int L  = (m + BLK - 1) / BLK;
  int lo = tid * L;
  int hi = imin(lo + L, m);
  int s = 0;
  for (int i = lo; i < hi; ++i) s += data[i];
  partial[tid] = s;
  __syncthreads();
  if (tid == 0) {
    int run = 0;
    for (int i = 0; i < BLK; ++i) { int v = partial[i]; partial[i] = run; run += v; }
  }
  __syncthreads();
  int run = partial[tid];
  for (int i = lo; i < hi; ++i) { int v = data[i]; data[i] = run; run += v; }
}

// --------------------------------------------- radix sort: stable scatter pass
__global__ void wl_radix_scatter(const u64* __restrict__ keyIn, const int* __restrict__ idxIn,
                                 u64* __restrict__ keyOut, int* __restrict__ idxOut,
                                 const int* __restrict__ hist, int nBlocks,
                                 int n, int shift) {
  __shared__ int base[256];     // scanned global offsets for (digit, this block)
  __shared__ int runHist[256];  // per-digit count of earlier tiles in this chunk
  __shared__ int tileDig[BLK];
  __shared__ int tileCnt[256];
  int tid = threadIdx.x;
  base[tid]    = hist[tid * nBlocks + blockIdx.x];
  runHist[tid] = 0;
  int chunkStart = blockIdx.x * CHUNK;
  for (int t = 0; t < TILES; ++t) {
    int j = chunkStart + t * BLK + tid;
    bool valid = (j < n);
    u64 key = 0; int dig = -1; int iv = 0;
    if (valid) {
      key = keyIn[j];
      dig = (int)((key >> shift) & 255ull);
      iv  = idxIn[j];
    }
    tileDig[tid] = dig;
    tileCnt[tid] = 0;
    __syncthreads();
    if (valid) {
      int r = 0;                                    // stable rank within tile
      for (int s = 0; s < tid; ++s) r += (tileDig[s] == dig) ? 1 : 0;
      int pos = base[dig] + runHist[dig] + r;
      keyOut[pos] = key;
      idxOut[pos] = iv;
      atomicAdd(&tileCnt[dig], 1);
    }
    __syncthreads();
    runHist[tid] += tileCnt[tid];
    __syncthreads();
  }
}

// ------------------------------------------------------- unique-rank relabel
__global__ void wl_mark_heads(const u64* __restrict__ keys, int* __restrict__ head, int n) {
  int i = blockIdx.x * blockDim.x + threadIdx.x;
  if (i < n) head[i] = (i == 0 || keys[i] != keys[i - 1]) ? 1 : 0;
}

__global__ void wl_chunk_sums(const int* __restrict__ head, int* __restrict__ sums, int n) {
  __shared__ int sh[BLK];
  int tid = threadIdx.x;
  int base = blockIdx.x * CHUNK;
  int s = 0;
  for (int t = 0; t < TILES; ++t) {
    int j = base + t * BLK + tid;
    if (j < n) s += head[j];
  }
  sh[tid] = s;
  __syncthreads();
  for (int off = BLK / 2; off > 0; off >>= 1) {
    if (tid < off) sh[tid] += sh[tid + off];
    __syncthreads();
  }
  if (tid == 0) sums[blockIdx.x] = sh[0];
}

// colors[idx[j]] = (inclusive #heads up to j) - 1  == rank among sorted distinct sigs
__global__ void wl_relabel(const int* __restrict__ head, const int* __restrict__ chunkBase,
                           const int* __restrict__ idx, i64* __restrict__ colors, int n) {
  __shared__ int sh[BLK];
  __shared__ int carry;
  int tid = threadIdx.x;
  if (tid == 0) carry = chunkBase[blockIdx.x];
  __syncthreads();
  int base = blockIdx.x * CHUNK;
  for (int t = 0; t < TILES; ++t) {
    int j = base + t * BLK + tid;
    int h = (j < n) ? head[j] : 0;
    sh[tid] = h;
    __syncthreads();
    for (int off = 1; off < BLK; off <<= 1) {         // Hillis-Steele inclusive scan
      int v = (tid >= off) ? sh[tid - off] : 0;
      __syncthreads();
      sh[tid] += v;
      __syncthreads();
    }
    int incl  = sh[tid];
    int total = sh[BLK - 1];
    if (j < n) colors[idx[j]] = (i64)(carry + incl - 1);
    __syncthreads();
    if (tid == 0) carry += total;
    __syncthreads();
  }
}

__global__ void wl_write_out(const i64* __restrict__ colors, float* __restrict__ out, int n) {
  int i = blockIdx.x * blockDim.x + threadIdx.x;
  if (i < n) out[i] = (float)colors[i];
}

// ------------------------------------------------------------------- launcher
extern "C" void kernel_launch(void* const* d_in, const int* in_sizes, int n_in,
                              void* d_out, int out_size, void* d_ws, size_t ws_size,
                              hipStream_t stream) {
  const i64* x    = (const i64*)d_in[0];
  const i64* edge = (const i64*)d_in[1];
  const i64* ht   = (const i64*)d_in[2];
  int       n = in_sizes[0];
  long long E = (long long)in_sizes[1] / 2;
  const i64* src = edge;
  const i64* dst = edge + E;

  // carve workspace (256B-aligned slices)
  unsigned char* w = (unsigned char*)d_ws;
  size_t pos = 0;
  auto take = [&](size_t bytes) -> void* {
    void* p = w + pos;
    pos += (bytes + 255) & ~(size_t)255;
    return p;
  };
  int nBlocks = (n + CHUNK - 1) / CHUNK;
  i64* colors = (i64*)take((size_t)n * 8);
  u64* msum   = (u64*)take((size_t)n * 8);
  u64* keyA   = (u64*)take((size_t)n * 8);
  u64* keyB   = (u64*)take((size_t)n * 8);
  int* idxA   = (int*)take((size_t)n * 4);
  int* idxB   = (int*)take((size_t)n * 4);
  int* head   = (int*)take((size_t)n * 4);
  int* hist   = (int*)take((size_t)256 * nBlocks * 4);
  int* csums  = (int*)take((size_t)nBlocks * 4);
  (void)ws_size; (void)n_in; (void)out_size;

  int gN = (n + BLK - 1) / BLK;
  long long nTiles = (E + TILE_E - 1) / TILE_E;
  long long gEcap  = nTiles < 8192 ? nTiles : 8192;
  int gE = (int)(gEcap > 0 ? gEcap : 1);
  // async b128 path requires 16B-aligned src/dst streams (dst = edge + E)
  int asyncOK = (((uintptr_t)src & 15) == 0 && ((uintptr_t)dst & 15) == 0) ? 1 : 0;

  wl_init_colors<<<gN, BLK, 0, stream>>>(x, colors, n);

  for (int it = 0; it < NUM_IT; ++it) {
    wl_clear_msum<<<gN, BLK, 0, stream>>>(msum, n);
    wl_edge_scatter<<<gE, BLK, 0, stream>>>(src, dst, colors, ht, msum, E, nTiles, asyncOK);
    wl_make_keys<<<gN, BLK, 0, stream>>>(colors, msum, keyA, idxA, n);

    // 8x8-bit stable LSD radix sort; result lands back in keyA/idxA
    for (int p = 0; p < 8; ++p) {
      const u64* ki = (p & 1) ? keyB : keyA;
      const int* ii = (p & 1) ? idxB : idxA;
      u64*       ko = (p & 1) ? keyA : keyB;
      int*       io = (p & 1) ? idxA : idxB;
      wl_radix_hist<<<nBlocks, BLK, 0, stream>>>(ki, n, 8 * p, hist, nBlocks);
      wl_scan_exclusive<<<1, BLK, 0, stream>>>(hist, 256 * nBlocks);
      wl_radix_scatter<<<nBlocks, BLK, 0, stream>>>(ki, ii, ko, io, hist, nBlocks, n, 8 * p);
    }

    wl_mark_heads<<<gN, BLK, 0, stream>>>(keyA, head, n);
    wl_chunk_sums<<<nBlocks, BLK, 0, stream>>>(head, csums, n);
    wl_scan_exclusive<<<1, BLK, 0, stream>>>(csums, nBlocks);
    wl_relabel<<<nBlocks, BLK, 0, stream>>>(head, csums, idxA, colors, n);
  }

  wl_write_out<<<gN, BLK, 0, stream>>>(colors, (float*)d_out, n);
}